// OptimizedIPGLayer_67164698575301
// MI455X (gfx1250) — compile-verified
//
#include <hip/hip_runtime.h>
#include <hip/hip_bf16.h>
#include <math.h>

typedef __attribute__((ext_vector_type(2))) float v2f;
typedef __attribute__((ext_vector_type(8))) float v8f;

#define B_  2
#define C_  128
#define H_  160
#define W_  160
#define HW  (H_ * W_)      // 25600
#define HD  80
#define WD  80
#define HWD (HD * WD)      // 6400
#define CH  256            // hidden channels (2*C)

// ---------------------------------------------------------------------------
// K1: 2x2 mean downsample  x[B,C,160,160] -> xd[B,C,80,80]
// ---------------------------------------------------------------------------
__global__ void k_down(const float* __restrict__ x, float* __restrict__ xd) {
    int idx = blockIdx.x * blockDim.x + threadIdx.x;           // B*C*6400
    if (idx >= B_ * C_ * HWD) return;
    int j  = idx % WD;
    int t  = idx / WD;
    int i  = t % HD;
    int bc = t / HD;
    const float* xp = x + (size_t)bc * HW;
    int r0 = (2 * i) * W_ + 2 * j;
    xd[idx] = 0.25f * (xp[r0] + xp[r0 + 1] + xp[r0 + W_] + xp[r0 + W_ + 1]);
}

// ---------------------------------------------------------------------------
// K2: df[b,h,w] = sum_c |x - bilinear_up(xd)|
// ---------------------------------------------------------------------------
__global__ void k_df(const float* __restrict__ x, const float* __restrict__ xd,
                     float* __restrict__ df) {
    int idx = blockIdx.x * blockDim.x + threadIdx.x;           // B*HW
    if (idx >= B_ * HW) return;
    int w = idx % W_;
    int h = (idx / W_) % H_;
    int b = idx / HW;

    float ch = fmaxf(0.5f * (float)h - 0.25f, 0.0f);
    int h0 = (int)ch; if (h0 > HD - 1) h0 = HD - 1;
    int h1 = h0 + 1;  if (h1 > HD - 1) h1 = HD - 1;
    float wh = ch - (float)h0;

    float cw = fmaxf(0.5f * (float)w - 0.25f, 0.0f);
    int w0 = (int)cw; if (w0 > WD - 1) w0 = WD - 1;
    int w1 = w0 + 1;  if (w1 > WD - 1) w1 = WD - 1;
    float ww = cw - (float)w0;

    const float* xb  = x  + (size_t)b * C_ * HW + h * W_ + w;
    const float* xdb = xd + (size_t)b * C_ * HWD;
    float acc = 0.0f;
    for (int c = 0; c < C_; ++c) {
        const float* d = xdb + c * HWD;
        float v00 = d[h0 * WD + w0], v01 = d[h0 * WD + w1];
        float v10 = d[h1 * WD + w0], v11 = d[h1 * WD + w1];
        float top = v00 * (1.0f - ww) + v01 * ww;
        float bot = v10 * (1.0f - ww) + v11 * ww;
        float xu  = top * (1.0f - wh) + bot * wh;
        acc += fabsf(xb[(size_t)c * HW] - xu);
    }
    df[idx] = acc;
}

// ---------------------------------------------------------------------------
// K3: per-batch min/max of df  (one block per batch)
// ---------------------------------------------------------------------------
__global__ void k_minmax(const float* __restrict__ df, float* __restrict__ mm) {
    __shared__ float smin[256], smax[256];
    int b = blockIdx.x;
    const float* d = df + (size_t)b * HW;
    float mn = 3.4e38f, mx = -3.4e38f;
    for (int i = threadIdx.x; i < HW; i += 256) {
        float v = d[i];
        mn = fminf(mn, v);
        mx = fmaxf(mx, v);
    }
    int t = threadIdx.x;
    smin[t] = mn; smax[t] = mx;
    __syncthreads();
    for (int s = 128; s > 0; s >>= 1) {
        if (t < s) { smin[t] = fminf(smin[t], smin[t + s]); smax[t] = fmaxf(smax[t], smax[t + s]); }
        __syncthreads();
    }
    if (t == 0) { mm[b * 2] = smin[0]; mm[b * 2 + 1] = smax[0]; }
}

// ---------------------------------------------------------------------------
// K4: per-pixel connection count k in [1,9]
// ---------------------------------------------------------------------------
__global__ void k_counts(const float* __restrict__ df, const float* __restrict__ mm,
                         int* __restrict__ kk) {
    int idx = blockIdx.x * blockDim.x + threadIdx.x;
    if (idx >= B_ * HW) return;
    int b = idx / HW;
    float mn = mm[b * 2], mx = mm[b * 2 + 1];
    float dfn = (df[idx] - mn) / (mx - mn + 1e-8f);
    float d2 = dfn * dfn;
    float dfp = d2 * d2;
    float cnt = 1.0f;
    const float thr = 0.9f;
    if (dfp > thr) {
        float above = rintf((dfp - thr) / (1.0f - thr + 1e-8f) * 15.0f);
        if (above < 0.0f) above = 0.0f;
        cnt = 1.0f + above;
    }
    int k = (int)cnt;
    if (k > 9) k = 9;
    kk[idx] = k;
}

// ---------------------------------------------------------------------------
// K5: GroupNorm stats per (b, g): mean and rsqrt(var+eps) over 4 ch x HW
// ---------------------------------------------------------------------------
__global__ void k_gnstats(const float* __restrict__ x, float* __restrict__ gn) {
    __shared__ float ss[256], ss2[256];
    int b = blockIdx.x >> 5;       // 0..1
    int g = blockIdx.x & 31;       // 0..31
    const float* xb = x + ((size_t)b * C_ + g * 4) * HW;  // 4 contiguous channels
    const int N = 4 * HW;          // 102400
    float s = 0.0f, s2 = 0.0f;
    for (int i = threadIdx.x; i < N; i += 256) {
        float v = xb[i];
        s += v; s2 += v * v;
    }
    int t = threadIdx.x;
    ss[t] = s; ss2[t] = s2;
    __syncthreads();
    for (int st = 128; st > 0; st >>= 1) {
        if (t < st) { ss[t] += ss[t + st]; ss2[t] += ss2[t + st]; }
        __syncthreads();
    }
    if (t == 0) {
        float mu  = ss[0] / (float)N;
        float var = ss2[0] / (float)N - mu * mu;
        gn[blockIdx.x * 2]     = mu;
        gn[blockIdx.x * 2 + 1] = rsqrtf(var + 1e-5f);
    }
}

// ---------------------------------------------------------------------------
// K6: cosine-sim top-k neighbor mixing + GroupNorm -> enhanced
// ---------------------------------------------------------------------------
__global__ void k_enhanced(const float* __restrict__ x, const int* __restrict__ kk,
                           const float* __restrict__ gn,
                           const float* __restrict__ gamma, const float* __restrict__ beta,
                           float* __restrict__ enh) {
    int idx = blockIdx.x * blockDim.x + threadIdx.x;
    if (idx >= B_ * HW) return;
    int w = idx % W_;
    int h = (idx / W_) % H_;
    int b = idx / HW;

    const float* xb = x + (size_t)b * C_ * HW + h * W_ + w;

    bool valid[9];
    int  off[9];
    {
        int j = 0;
        for (int dy = -1; dy <= 1; ++dy)
            for (int dx = -1; dx <= 1; ++dx, ++j) {
                int hh = h + dy, wv = w + dx;
                valid[j] = (hh >= 0) && (hh < H_) && (wv >= 0) && (wv < W_);
                off[j]   = dy * W_ + dx;
            }
    }

    float dot[9], pn[9];
    #pragma unroll
    for (int j = 0; j < 9; ++j) { dot[j] = 0.0f; pn[j] = 0.0f; }
    float cn = 0.0f;

    for (int c = 0; c < C_; ++c) {
        const float* p = xb + (size_t)c * HW;
        float cv = *p;
        cn += cv * cv;
        #pragma unroll
        for (int j = 0; j < 9; ++j) {
            float pv = valid[j] ? p[off[j]] : 0.0f;
            dot[j] += cv * pv;
            pn[j]  += pv * pv;
        }
    }

    float cnorm = fmaxf(sqrtf(cn), 1e-12f);
    float sim[9];
    #pragma unroll
    for (int j = 0; j < 9; ++j)
        sim[j] = dot[j] / (cnorm * fmaxf(sqrtf(pn[j]), 1e-12f));

    // stable descending selection (matches stable argsort(-sim))
    int order[9];
    bool used[9];
    #pragma unroll
    for (int j = 0; j < 9; ++j) used[j] = false;
    for (int r = 0; r < 9; ++r) {
        int best = 0; float bv = -1e30f; bool found = false;
        for (int j = 0; j < 9; ++j) {
            if (!used[j] && (!found || sim[j] > bv)) { bv = sim[j]; best = j; found = true; }
        }
        used[best] = true;
        order[r] = best;
    }

    int kcnt = kk[idx];
    float wj[9];
    #pragma unroll
    for (int j = 0; j < 9; ++j) wj[j] = 0.0f;
    float wsum = 0.0f;
    for (int r = 0; r < 9; ++r) {
        if (r < kcnt) {
            float e = expf(sim[order[r]]);
            wj[order[r]] = e;
            wsum += e;
        }
    }
    float inv = 1.0f / (wsum + 1e-8f);

    const float* gnb = gn + b * 64;
    for (int c = 0; c < C_; ++c) {
        const float* p = xb + (size_t)c * HW;
        float o = 0.0f;
        #pragma unroll
        for (int j = 0; j < 9; ++j) {
            float pv = valid[j] ? p[off[j]] : 0.0f;
            o += wj[j] * pv;
        }
        o *= inv;
        int g = c >> 2;
        float mu = gnb[g * 2], rs = gnb[g * 2 + 1];
        float nx = (*p - mu) * rs * gamma[c] + beta[c];
        enh[((size_t)b * C_ + c) * HW + h * W_ + w] = o + nx;
    }
}

// ---------------------------------------------------------------------------
// K7: GEMM1 via fp32 WMMA, 32x32 register-blocked tile per wave:
//     hbuf[256,P] = relu(w1[256x128] @ E[128,P] + b1)
// Per K-step: 2 A fragments + 2 B fragments -> 4x V_WMMA_F32_16X16X4_F32.
// ---------------------------------------------------------------------------
__global__ void k_ffn_gemm1(const float* __restrict__ E, const float* __restrict__ w1,
                            const float* __restrict__ b1, float* __restrict__ hbuf) {
    const int NT = HW / 32;                                    // 800 N tiles (32-wide)
    int wave = (blockIdx.x * blockDim.x + threadIdx.x) >> 5;
    int lane = threadIdx.x & 31;
    int ntile = wave % NT;
    int rest  = wave / NT;
    int mtile = rest % (CH / 32);                              // 8 M tiles (32-tall)
    int b     = rest / (CH / 32);

    int lo   = lane & 15;
    int half = lane >> 4;
    int m0   = mtile * 32 + lo;          // A fragment rows, tile 0
    int m1   = m0 + 16;                  // tile 1
    int n0   = ntile * 32 + lo;          // B fragment cols, tile 0
    int n1   = n0 + 16;                  // tile 1

    const float* Eb = E + (size_t)b * C_ * HW;
    v8f a00 = {}, a01 = {}, a10 = {}, a11 = {};   // acc[mi][ni]
    #pragma unroll 4
    for (int kt = 0; kt < C_ / 4; ++kt) {                      // 32 K-steps
        int k = kt * 4 + half * 2;
        v2f A0, A1, B0, B1;
        A0.x = w1[m0 * C_ + k];     A0.y = w1[m0 * C_ + k + 1];
        A1.x = w1[m1 * C_ + k];     A1.y = w1[m1 * C_ + k + 1];
        B0.x = Eb[(size_t)k * HW + n0];
        B0.y = Eb[(size_t)(k + 1) * HW + n0];
        B1.x = Eb[(size_t)k * HW + n1];
        B1.y = Eb[(size_t)(k + 1) * HW + n1];
        a00 = __builtin_amdgcn_wmma_f32_16x16x4_f32(false, A0, false, B0, (short)0, a00, false, false);
        a01 = __builtin_amdgcn_wmma_f32_16x16x4_f32(false, A0, false, B1, (short)0, a01, false, false);
        a10 = __builtin_amdgcn_wmma_f32_16x16x4_f32(false, A1, false, B0, (short)0, a10, false, false);
        a11 = __builtin_amdgcn_wmma_f32_16x16x4_f32(false, A1, false, B1, (short)0, a11, false, false);
    }

    float* Hb = hbuf + (size_t)b * CH * HW;
    #pragma unroll
    for (int mi = 0; mi < 2; ++mi) {
        int mbase = mtile * 32 + mi * 16 + half * 8;
        #pragma unroll
        for (int r = 0; r < 8; ++r) {
            int m = mbase + r;
            float bias = b1[m];
            float v0 = (mi == 0 ? a00[r] : a10[r]) + bias;
            float v1 = (mi == 0 ? a01[r] : a11[r]) + bias;
            Hb[(size_t)m * HW + n0] = v0 > 0.0f ? v0 : 0.0f;
            Hb[(size_t)m * HW + n1] = v1 > 0.0f ? v1 : 0.0f;
        }
    }
}

// ---------------------------------------------------------------------------
// K8: GEMM2 via fp32 WMMA, 32x32 register-blocked tile per wave:
//     out[128,P] = E + w2[128x256] @ hbuf + b2
// ---------------------------------------------------------------------------
__global__ void k_ffn_gemm2(const float* __restrict__ hbuf, const float* __restrict__ w2,
                            const float* __restrict__ b2, const float* __restrict__ E,
                            float* __restrict__ out) {
    const int NT = HW / 32;                                    // 800
    int wave = (blockIdx.x * blockDim.x + threadIdx.x) >> 5;
    int lane = threadIdx.x & 31;
    int ntile = wave % NT;
    int rest  = wave / NT;
    int mtile = rest % (C_ / 32);                              // 4 M tiles (32-tall)
    int b     = rest / (C_ / 32);

    int lo   = lane & 15;
    int half = lane >> 4;
    int m0   = mtile * 32 + lo;
    int m1   = m0 + 16;
    int n0   = ntile * 32 + lo;
    int n1   = n0 + 16;

    const float* Hb = hbuf + (size_t)b * CH * HW;
    v8f a00 = {}, a01 = {}, a10 = {}, a11 = {};
    #pragma unroll 4
    for (int kt = 0; kt < CH / 4; ++kt) {                      // 64 K-steps
        int k = kt * 4 + half * 2;
        v2f A0, A1, B0, B1;
        A0.x = w2[m0 * CH + k];     A0.y = w2[m0 * CH + k + 1];
        A1.x = w2[m1 * CH + k];     A1.y = w2[m1 * CH + k + 1];
        B0.x = Hb[(size_t)k * HW + n0];
        B0.y = Hb[(size_t)(k + 1) * HW + n0];
        B1.x = Hb[(size_t)k * HW + n1];
        B1.y = Hb[(size_t)(k + 1) * HW + n1];
        a00 = __builtin_amdgcn_wmma_f32_16x16x4_f32(false, A0, false, B0, (short)0, a00, false, false);
        a01 = __builtin_amdgcn_wmma_f32_16x16x4_f32(false, A0, false, B1, (short)0, a01, false, false);
        a10 = __builtin_amdgcn_wmma_f32_16x16x4_f32(false, A1, false, B0, (short)0, a10, false, false);
        a11 = __builtin_amdgcn_wmma_f32_16x16x4_f32(false, A1, false, B1, (short)0, a11, false, false);
    }

    const float* Eb = E + (size_t)b * C_ * HW;
    float* Ob = out + (size_t)b * C_ * HW;
    #pragma unroll
    for (int mi = 0; mi < 2; ++mi) {
        int mbase = mtile * 32 + mi * 16 + half * 8;
        #pragma unroll
        for (int r = 0; r < 8; ++r) {
            int m = mbase + r;
            float bias = b2[m];
            size_t o0 = (size_t)m * HW + n0;
            size_t o1 = (size_t)m * HW + n1;
            float v0 = (mi == 0 ? a00[r] : a10[r]);
            float v1 = (mi == 0 ? a01[r] : a11[r]);
            Ob[o0] = Eb[o0] + v0 + bias;
            Ob[o1] = Eb[o1] + v1 + bias;
        }
    }
}

// ---------------------------------------------------------------------------
// Host launch
// ---------------------------------------------------------------------------
extern "C" void kernel_launch(void* const* d_in, const int* in_sizes, int n_in,
                              void* d_out, int out_size, void* d_ws, size_t ws_size,
                              hipStream_t stream) {
    (void)in_sizes; (void)n_in; (void)out_size; (void)ws_size;
    const float* x     = (const float*)d_in[0];
    const float* gamma = (const float*)d_in[1];
    const float* beta  = (const float*)d_in[2];
    const float* w1    = (const float*)d_in[3];
    const float* b1    = (const float*)d_in[4];
    const float* w2    = (const float*)d_in[5];
    const float* b2    = (const float*)d_in[6];
    float* out = (float*)d_out;

    // workspace layout
    char* ws = (char*)d_ws;
    size_t off = 0;
    float* xd  = (float*)(ws + off); off += (size_t)B_ * C_ * HWD * sizeof(float);   // 6.55 MB
    float* df  = (float*)(ws + off); off += (size_t)B_ * HW * sizeof(float);         // 205 KB
    float* mm  = (float*)(ws + off); off += 16;                                       // B*2 floats
    int*   kk  = (int*)  (ws + off); off += (size_t)B_ * HW * sizeof(int);           // 205 KB
    float* gn  = (float*)(ws + off); off += (size_t)B_ * 32 * 2 * sizeof(float);     // 512 B
    off = (off + 255) & ~(size_t)255;
    float* enh = (float*)(ws + off); off += (size_t)B_ * C_ * HW * sizeof(float);    // 26.2 MB
    float* hb  = (float*)(ws + off); off += (size_t)B_ * CH * HW * sizeof(float);    // 52.4 MB

    const int T = 256;

    // K1: downsample
    k_down<<<(B_ * C_ * HWD + T - 1) / T, T, 0, stream>>>(x, xd);
    // K2: df
    k_df<<<(B_ * HW + T - 1) / T, T, 0, stream>>>(x, xd, df);
    // K3: per-batch min/max
    k_minmax<<<B_, T, 0, stream>>>(df, mm);
    // K4: counts
    k_counts<<<(B_ * HW + T - 1) / T, T, 0, stream>>>(df, mm, kk);
    // K5: GroupNorm stats
    k_gnstats<<<B_ * 32, T, 0, stream>>>(x, gn);
    // K6: enhanced = top-k neighbor mix + GroupNorm
    k_enhanced<<<(B_ * HW + T - 1) / T, T, 0, stream>>>(x, kk, gn, gamma, beta, enh);
    // K7: FFN GEMM1 (WMMA, 32x32 tiles): 2*8*800 = 12800 waves, 8 waves/block
    {
        int waves = B_ * (CH / 32) * (HW / 32);
        k_ffn_gemm1<<<waves / 8, T, 0, stream>>>(enh, w1, b1, hb);
    }
    // K8: FFN GEMM2 (WMMA, 32x32 tiles) + residual: 2*4*800 = 6400 waves
    {
        int waves = B_ * (C_ / 32) * (HW / 32);
        k_ffn_gemm2<<<waves / 8, T, 0, stream>>>(hb, w2, b2, enh, out);
    }
}